// IRadon_644245094842
// MI455X (gfx1250) — compile-verified
//
#include <hip/hip_runtime.h>
#include <math.h>

// Problem constants (from reference)
#define W      384      // detector bins == image side
#define NANG   180      // angles
#define NPAD   1024     // FFT pad size: 2^ceil(log2(2*384))
#define NBATCH 2
#define HEXT   768      // extended/reversed circulant tap table size

// Workspace layout (float offsets into d_ws)
#define OFF_H    0                      // h[1024] ramp-filter impulse response
#define OFF_COS  1024                   // cos(theta_l), 180
#define OFF_SIN  1216                   // sin(theta_l), 180
#define OFF_FT   1536                   // filtT[2][180][384] angle-major filtered sinogram

typedef __attribute__((ext_vector_type(2))) float v2f;
typedef __attribute__((ext_vector_type(8))) float v8f;

// ---- gfx1250 async global->LDS copy (guarded so the file always compiles) ----
#if __has_builtin(__builtin_amdgcn_global_load_async_to_lds_b32) && \
    __has_builtin(__builtin_amdgcn_s_wait_asynccnt)
#define USE_ASYNC_LDS 1
typedef __attribute__((address_space(1))) int gint_t;   // builtin wants global int*
typedef __attribute__((address_space(3))) int lint_t;   // LDS int*
__device__ __forceinline__ void async_cp_f32(const float* g, float* l) {
    __builtin_amdgcn_global_load_async_to_lds_b32(
        (gint_t*)g, (lint_t*)l, /*imm offset=*/0, /*cpol=*/0);
}
__device__ __forceinline__ void async_wait0() {
    __builtin_amdgcn_s_wait_asynccnt(0);
}
#else
#define USE_ASYNC_LDS 0
__device__ __forceinline__ void async_cp_f32(const float* g, float* l) { *l = *g; }
__device__ __forceinline__ void async_wait0() {}
#endif

// ---------------------------------------------------------------------------
// Kernel 1: build ramp-filter impulse response h[n] = (1/N) sum_k filt[k]*cos(2pi k n/N)
// (real ifft of the real, even 2|f| ramp), plus per-angle cos/sin tables.
// Exact phase reduction via (k*n) & (N-1) keeps cosf args in [0, 2pi).
// ---------------------------------------------------------------------------
__global__ __launch_bounds__(1024) void ir_init(float* __restrict__ ws) {
    const int n = threadIdx.x;                 // 0..1023
    const float step = 6.283185307179586f / (float)NPAD;
    float s = 0.0f;
    for (int k = 0; k < NPAD; ++k) {
        int kk = k < (NPAD - k) ? k : (NPAD - k);      // min(k, N-k) == |fftfreq|*N
        float filt = 2.0f * (float)kk / (float)NPAD;   // 2*|f_k|
        int m = (k * n) & (NPAD - 1);
        s = fmaf(filt, cosf((float)m * step), s);
    }
    ws[OFF_H + n] = s * (1.0f / (float)NPAD);
    if (n < NANG) {
        float rad = (float)n * (3.14159265358979323846f / 180.0f);
        ws[OFF_COS + n] = cosf(rad);
        ws[OFF_SIN + n] = sinf(rad);
    }
}

// ---------------------------------------------------------------------------
// Kernel 2: ramp filter as fp32 WMMA GEMM.
//   y[b,i,l] = sum_j H[i,j] * x[b,0,j,l],  H[i,j] = h[(i-j) & 1023]
// Block = (batch, N-tile): stages the extended+reversed tap table
//   g[t] = h[(383 - t) & 1023]   (so A taps are g[qbase+ka], g[qbase+ka+1]
//   with qbase = 383 - rowA: linear LDS addressing, no masking in the loop)
// and the zero-padded 384x16 sinogram panel into LDS, then 8 waves x 3
// M-tiles each run a pure ds_load + V_WMMA_F32_16X16X4_F32 inner loop
// (K=384, 96 steps/tile). Output written angle-major: filtT[b][l][i].
// ---------------------------------------------------------------------------
__global__ __launch_bounds__(256) void ir_filt_gemm(const float* __restrict__ x,
                                                    const float* __restrict__ ws) {
    __shared__ float shHe[HEXT];        // extended reversed circulant taps (3 KB)
    __shared__ float shB[W * 16];       // B panel [k][n], zero-padded cols (24 KB)

    const int batch = blockIdx.x / 12;
    const int nt    = blockIdx.x % 12;
    const int l0    = nt * 16;
    const float* __restrict__ xb = x + batch * (W * NANG);

    // Stage extended reversed tap table: shHe[t] = h[(383 - t) & 1023]
    for (int t = threadIdx.x; t < HEXT; t += 256)
        async_cp_f32(&ws[OFF_H + ((383 - t) & (NPAD - 1))], &shHe[t]);
    // Stage B panel: shB[k*16 + c] = x[k][l0+c] (0 beyond the 180 valid angles)
    for (int t = threadIdx.x; t < W * 16; t += 256) {
        const int k = t >> 4;
        const int c = t & 15;
        const int l = l0 + c;
        if (l < NANG) async_cp_f32(&xb[k * NANG + l], &shB[t]);
        else          shB[t] = 0.0f;
    }
    async_wait0();
    __syncthreads();

    const int wave = threadIdx.x >> 5;
    const int lane = threadIdx.x & 31;
    const int half = lane >> 4;                // K sub-pair select (ISA 16x4 f32 layout)
    const int mn   = lane & 15;                // M row (A) / N col (B, D)
    const int col  = l0 + mn;
    const bool colok = (col < NANG);
    float* __restrict__ ftcol = (float*)ws + OFF_FT + (batch * NANG + col) * W;
    const float* __restrict__ bbase = &shB[mn];

    for (int sub = 0; sub < 3; ++sub) {        // 8 waves x 3 = 24 M-tiles
        const int mt    = wave * 3 + sub;
        const int i0    = mt * 16;
        const int rowA  = i0 + mn;
        const float* __restrict__ hbase = &shHe[(W - 1) - rowA];  // qbase

        v8f c = {0.f, 0.f, 0.f, 0.f, 0.f, 0.f, 0.f, 0.f};
        for (int k0 = 0; k0 < W; k0 += 4) {
            const int ka = k0 + half * 2;
            v2f a, b;
            a.x = hbase[ka];                   // h[(rowA-ka)   & 1023]
            a.y = hbase[ka + 1];               // h[(rowA-ka-1) & 1023]
            b.x = bbase[ ka      * 16];
            b.y = bbase[(ka + 1) * 16];
            c = __builtin_amdgcn_wmma_f32_16x16x4_f32(
                    /*neg_a=*/false, a, /*neg_b=*/false, b,
                    /*c_mod=*/(short)0, c, /*reuse_a=*/false, /*reuse_b=*/false);
        }

        // D layout: VGPR v holds row (half*8 + v), lane column = mn.
        if (colok) {
            const int ibase = i0 + half * 8;
#pragma unroll
            for (int v = 0; v < 8; ++v)
                ftcol[ibase + v] = c[v];
        }
    }
}

// ---------------------------------------------------------------------------
// Kernel 3: backprojection. ix == l exactly (linspace over NANG == W_in points),
// so only 1-D bilinear sampling along the detector axis remains.
// Each block owns a 16x16 output tile; per angle, the contiguous 384-float
// filtered column is async-staged into LDS by all 256 threads.
// ---------------------------------------------------------------------------
__global__ __launch_bounds__(256) void ir_backproj(const float* __restrict__ ws,
                                                   float* __restrict__ out) {
    __shared__ float colbuf[W];
    __shared__ float ct[NANG], st[NANG];

    const int tx = threadIdx.x & 15;
    const int ty = threadIdx.x >> 4;
    const int j = blockIdx.x * 16 + tx;        // image x (fast axis)
    const int i = blockIdx.y * 16 + ty;        // image y
    const int batch = blockIdx.z;

    for (int t = threadIdx.x; t < NANG; t += 256) {
        ct[t] = ws[OFF_COS + t];
        st[t] = ws[OFF_SIN + t];
    }

    const float uj = -1.0f + 2.0f * (float)j / 383.0f;   // xgrid
    const float ui = -1.0f + 2.0f * (float)i / 383.0f;   // ygrid
    const float* __restrict__ ft = ws + OFF_FT + batch * (NANG * W);

    float acc = 0.0f;
    for (int l = 0; l < NANG; ++l) {
        __syncthreads();                                   // colbuf readers done
        for (int t = threadIdx.x; t < W; t += 256)
            async_cp_f32(&ft[l * W + t], &colbuf[t]);
        if (l + 1 < NANG)
            __builtin_prefetch(&ft[(l + 1) * W + threadIdx.x], 0, 1);
        async_wait0();
        __syncthreads();

        const float gy = uj * ct[l] - ui * st[l];
        const float t  = (gy + 1.0f) * 0.5f * 383.0f;      // detector coordinate
        const float f0 = floorf(t);
        const float w1 = t - f0;
        const int  p0  = (int)f0;
        float v0 = (p0 >= 0     && p0 <= W - 1)     ? colbuf[p0]     : 0.0f;
        float v1 = (p0 + 1 >= 0 && p0 + 1 <= W - 1) ? colbuf[p0 + 1] : 0.0f;
        acc = fmaf(v0, 1.0f - w1, acc);
        acc = fmaf(v1, w1, acc);
    }

    const bool circle = (uj * uj + ui * ui) <= 1.0f;
    const float scale = 3.14159265358979323846f / (2.0f * (float)NANG);
    out[batch * (W * W) + i * W + j] = circle ? acc * scale : 0.0f;
}

// ---------------------------------------------------------------------------
extern "C" void kernel_launch(void* const* d_in, const int* in_sizes, int n_in,
                              void* d_out, int out_size, void* d_ws, size_t ws_size,
                              hipStream_t stream) {
    const float* x = (const float*)d_in[0];    // (2,1,384,180) fp32
    float* ws  = (float*)d_ws;
    float* out = (float*)d_out;                // (2,1,384,384) fp32

    ir_init<<<1, 1024, 0, stream>>>(ws);
    ir_filt_gemm<<<12 * NBATCH, 256, 0, stream>>>(x, ws);
    dim3 grid(W / 16, W / 16, NBATCH);
    ir_backproj<<<grid, 256, 0, stream>>>(ws, out);
}